// GCN_19318762897897
// MI455X (gfx1250) — compile-verified
//
#include <hip/hip_runtime.h>
#include <math.h>

typedef __attribute__((ext_vector_type(2))) float v2f;
typedef __attribute__((ext_vector_type(8))) float v8f;

#define FEAT_IN 128

// ---------------------------------------------------------------------------
// Graph normalization setup
// ---------------------------------------------------------------------------
__global__ void k_init_deg(float* __restrict__ deg, int n) {
    int i = blockIdx.x * blockDim.x + threadIdx.x;
    if (i < n) deg[i] = 1.0f;  // self-loop contributes 1 to degree
}

__global__ void k_edge_prep(const long long* __restrict__ src64,
                            const long long* __restrict__ dst64,
                            int* __restrict__ src32, int* __restrict__ dst32,
                            float* __restrict__ deg, int e) {
    int i = blockIdx.x * blockDim.x + threadIdx.x;
    if (i < e) {
        int s = (int)__builtin_nontemporal_load(&src64[i]);
        int d = (int)__builtin_nontemporal_load(&dst64[i]);
        src32[i] = s;
        dst32[i] = d;
        atomicAdd(&deg[d], 1.0f);
    }
}

__global__ void k_rsqrt_inplace(float* __restrict__ deg, int n) {
    int i = blockIdx.x * blockDim.x + threadIdx.x;
    if (i < n) deg[i] = rsqrtf(deg[i]);  // deg >= 1 always (self loop)
}

__global__ void k_edge_norm(const int* __restrict__ src32,
                            const int* __restrict__ dst32,
                            const float* __restrict__ dinv,
                            float* __restrict__ norm, int e) {
    int i = blockIdx.x * blockDim.x + threadIdx.x;
    if (i < e) {
        int s = __builtin_nontemporal_load(&src32[i]);
        int d = __builtin_nontemporal_load(&dst32[i]);
        norm[i] = dinv[s] * dinv[d];
    }
}

// ---------------------------------------------------------------------------
// Layer-1 GEMM: X[N,128] @ W[128,16] -> Y[N,16] via V_WMMA_F32_16X16X4_F32.
// One wave per 16-row tile; 32 k-steps of 16x16x4. Full fp32 precision.
// W1 (8 KB) is staged once per block into LDS; per-step B fragments come from
// ds_load instead of repeated global loads.
// A (16x4 f32): lane L holds row (L&15), K-pair (L>>4)*2 in its 2 VGPRs.
// B (4x16 f32): lane L holds col (L&15), K-pair (L>>4)*2 in its 2 VGPRs.
// C/D (16x16 f32, 8 VGPRs): lane<16 -> rows 0..7, lane>=16 -> rows 8..15.
// ---------------------------------------------------------------------------
__global__ void k_gemm_wmma_128x16(const float* __restrict__ X,
                                   const float* __restrict__ W,
                                   float* __restrict__ Y, int n) {
    __shared__ float sW[FEAT_IN * 16];  // 8 KB

    // Cooperative stage of W into LDS: 512 float4 / 256 threads = 2 each.
    {
        const float4* Wv  = (const float4*)W;
        float4*       sWv = (float4*)sW;
        for (int t = threadIdx.x; t < (FEAT_IN * 16) / 4; t += blockDim.x)
            sWv[t] = Wv[t];
    }
    __syncthreads();  // before any wave-level early exit

    int gtid = blockIdx.x * blockDim.x + threadIdx.x;
    int wave = gtid >> 5;          // wave-uniform
    int lane = threadIdx.x & 31;
    int row0 = wave * 16;          // wave-uniform
    if (row0 >= n) return;         // whole-wave exit: EXEC full for WMMA

    int r  = lane & 15;            // A row within tile / B column
    int kb = (lane >> 4) << 1;     // 0 or 2: which K-pair this half-wave owns

    int row = row0 + r;
    if (row >= n) row = n - 1;     // clamp reads in tail tile (writes guarded)
    const float* xrow = X + (size_t)row * FEAT_IN;

    v8f c = {0.f, 0.f, 0.f, 0.f, 0.f, 0.f, 0.f, 0.f};
    #pragma unroll 4
    for (int k0 = 0; k0 < FEAT_IN; k0 += 4) {
        v2f a, b;
        a.x = xrow[k0 + kb];
        a.y = xrow[k0 + kb + 1];
        b.x = sW[(k0 + kb) * 16 + r];
        b.y = sW[(k0 + kb + 1) * 16 + r];
        c = __builtin_amdgcn_wmma_f32_16x16x4_f32(
                /*neg_a=*/false, a, /*neg_b=*/false, b,
                /*c_mod=*/(short)0, c, /*reuse_a=*/false, /*reuse_b=*/false);
    }

    int colw  = lane & 15;
    int rbase = row0 + ((lane >> 4) << 3);  // lanes>=16 hold rows 8..15
    if (row0 + 16 <= n) {
        // Fast path (every tile when n % 16 == 0): no per-element guards.
        #pragma unroll
        for (int v = 0; v < 8; ++v)
            Y[(size_t)(rbase + v) * 16 + colw] = c[v];
    } else {
        #pragma unroll
        for (int v = 0; v < 8; ++v) {
            int rr = rbase + v;
            if (rr < n) Y[(size_t)rr * 16 + colw] = c[v];
        }
    }
}

// ---------------------------------------------------------------------------
// Small per-node GEMMs (F tiny: negligible FLOPs, VALU is optimal)
// ---------------------------------------------------------------------------
template <int FI, int FO>
__global__ void k_gemm_small(const float* __restrict__ X,
                             const float* __restrict__ W,
                             float* __restrict__ Y, int n) {
    int i = blockIdx.x * blockDim.x + threadIdx.x;
    if (i >= n) return;
    float x[FI];
    #pragma unroll
    for (int k = 0; k < FI; ++k) x[k] = X[(size_t)i * FI + k];
    #pragma unroll
    for (int o = 0; o < FO; ++o) {
        float acc = 0.f;
        #pragma unroll
        for (int k = 0; k < FI; ++k) acc = fmaf(x[k], W[k * FO + o], acc);
        Y[(size_t)i * FO + o] = acc;
    }
}

// acc[i] = xw[i] * dinv[i]^2   (self-loop term; also initializes accumulator)
template <int F>
__global__ void k_self_init(const float* __restrict__ xw,
                            const float* __restrict__ dinv,
                            float* __restrict__ acc, int n) {
    int i = blockIdx.x * blockDim.x + threadIdx.x;
    if (i >= n) return;
    float w = dinv[i];
    w *= w;
    #pragma unroll
    for (int f = 0; f < F; ++f)
        acc[(size_t)i * F + f] = xw[(size_t)i * F + f] * w;
}

// Edge scatter, F=16: 4 threads/edge, each float4 gather + 4 float atomics
__global__ void k_scatter16(const int* __restrict__ src,
                            const int* __restrict__ dst,
                            const float* __restrict__ norm,
                            const float* __restrict__ xw,
                            float* __restrict__ acc, int e) {
    int t = blockIdx.x * blockDim.x + threadIdx.x;
    int i = t >> 2;
    if (i >= e) return;
    int   q = t & 3;
    int   s = __builtin_nontemporal_load(&src[i]);
    int   d = __builtin_nontemporal_load(&dst[i]);
    float w = __builtin_nontemporal_load(&norm[i]);
    const float4 v = *(const float4*)(xw + (size_t)s * 16 + q * 4);
    float* a = acc + (size_t)d * 16 + q * 4;
    atomicAdd(a + 0, v.x * w);
    atomicAdd(a + 1, v.y * w);
    atomicAdd(a + 2, v.z * w);
    atomicAdd(a + 3, v.w * w);
}

// Edge scatter, small F: 1 thread/edge
template <int F>
__global__ void k_scatter(const int* __restrict__ src,
                          const int* __restrict__ dst,
                          const float* __restrict__ norm,
                          const float* __restrict__ xw,
                          float* __restrict__ acc, int e) {
    int i = blockIdx.x * blockDim.x + threadIdx.x;
    if (i >= e) return;
    int   s = __builtin_nontemporal_load(&src[i]);
    int   d = __builtin_nontemporal_load(&dst[i]);
    float w = __builtin_nontemporal_load(&norm[i]);
    #pragma unroll
    for (int f = 0; f < F; ++f)
        atomicAdd(&acc[(size_t)d * F + f], xw[(size_t)s * F + f] * w);
}

template <int F>
__global__ void k_bias_tanh(const float* __restrict__ acc,
                            const float* __restrict__ b,
                            float* __restrict__ out, int n) {
    int i = blockIdx.x * blockDim.x + threadIdx.x;
    if (i >= n) return;
    #pragma unroll
    for (int f = 0; f < F; ++f)
        out[(size_t)i * F + f] = tanhf(acc[(size_t)i * F + f] + b[f]);
}

// out[i,c] = h3[i,0]*Wc[0,c] + h3[i,1]*Wc[1,c] + bc[c]
__global__ void k_classifier(const float* __restrict__ h3,
                             const float* __restrict__ Wc,
                             const float* __restrict__ bc,
                             float* __restrict__ out, int n) {
    int i = blockIdx.x * blockDim.x + threadIdx.x;
    if (i >= n) return;
    float a0 = h3[(size_t)i * 2 + 0];
    float a1 = h3[(size_t)i * 2 + 1];
    #pragma unroll
    for (int c = 0; c < 8; ++c)
        out[(size_t)i * 8 + c] = fmaf(a0, Wc[c], fmaf(a1, Wc[8 + c], bc[c]));
}

// ---------------------------------------------------------------------------
static inline size_t align256(size_t x) { return (x + 255) & ~(size_t)255; }

extern "C" void kernel_launch(void* const* d_in, const int* in_sizes, int n_in,
                              void* d_out, int out_size, void* d_ws, size_t ws_size,
                              hipStream_t stream) {
    const float*     h   = (const float*)d_in[0];
    const long long* ei  = (const long long*)d_in[1];   // int64 [2, E]
    const float*     W1  = (const float*)d_in[2];
    const float*     b1  = (const float*)d_in[3];
    const float*     W2  = (const float*)d_in[4];
    const float*     b2  = (const float*)d_in[5];
    const float*     W3  = (const float*)d_in[6];
    const float*     b3  = (const float*)d_in[7];
    const float*     Wc  = (const float*)d_in[8];
    const float*     bc  = (const float*)d_in[9];
    float*           out = (float*)d_out;

    const int N = in_sizes[0] / FEAT_IN;   // 100000
    const int E = in_sizes[1] / 2;         // 3200000
    const long long* src64 = ei;
    const long long* dst64 = ei + E;

    // Workspace layout
    char* w = (char*)d_ws;
    int*   src32 = (int*)w;   w += align256((size_t)E * sizeof(int));
    int*   dst32 = (int*)w;   w += align256((size_t)E * sizeof(int));
    float* norm  = (float*)w; w += align256((size_t)E * sizeof(float));
    float* dinv  = (float*)w; w += align256((size_t)N * sizeof(float)); // deg -> dinv
    float* bufA  = (float*)w; w += align256((size_t)N * 16 * sizeof(float)); // xw
    float* bufB  = (float*)w; w += align256((size_t)N * 16 * sizeof(float)); // acc
    float* bufC  = (float*)w; w += align256((size_t)N * 16 * sizeof(float)); // h1/h2

    float* h3 = out + (size_t)N * 8;  // second output region

    const int BLK   = 256;
    const int gN    = (N + BLK - 1) / BLK;
    const int gE    = (E + BLK - 1) / BLK;
    const int gE4   = (E * 4 + BLK - 1) / BLK;
    const int waves = (N + 15) / 16;                      // one wave per 16 rows
    const int gWmma = (waves * 32 + BLK - 1) / BLK;

    // --- normalization ---
    k_init_deg<<<gN, BLK, 0, stream>>>(dinv, N);
    k_edge_prep<<<gE, BLK, 0, stream>>>(src64, dst64, src32, dst32, dinv, E);
    k_rsqrt_inplace<<<gN, BLK, 0, stream>>>(dinv, N);
    k_edge_norm<<<gE, BLK, 0, stream>>>(src32, dst32, dinv, norm, E);

    // --- layer 1 (128 -> 16), GEMM via V_WMMA_F32_16X16X4_F32 ---
    k_gemm_wmma_128x16<<<gWmma, BLK, 0, stream>>>(h, W1, bufA, N);
    k_self_init<16><<<gN, BLK, 0, stream>>>(bufA, dinv, bufB, N);
    k_scatter16<<<gE4, BLK, 0, stream>>>(src32, dst32, norm, bufA, bufB, E);
    k_bias_tanh<16><<<gN, BLK, 0, stream>>>(bufB, b1, bufC, N);   // h1

    // --- layer 2 (16 -> 4) ---
    k_gemm_small<16, 4><<<gN, BLK, 0, stream>>>(bufC, W2, bufA, N);
    k_self_init<4><<<gN, BLK, 0, stream>>>(bufA, dinv, bufB, N);
    k_scatter<4><<<gE, BLK, 0, stream>>>(src32, dst32, norm, bufA, bufB, E);
    k_bias_tanh<4><<<gN, BLK, 0, stream>>>(bufB, b2, bufC, N);    // h2

    // --- layer 3 (4 -> 2) ---
    k_gemm_small<4, 2><<<gN, BLK, 0, stream>>>(bufC, W3, bufA, N);
    k_self_init<2><<<gN, BLK, 0, stream>>>(bufA, dinv, bufB, N);
    k_scatter<2><<<gE, BLK, 0, stream>>>(src32, dst32, norm, bufA, bufB, E);
    k_bias_tanh<2><<<gN, BLK, 0, stream>>>(bufB, b3, h3, N);      // h3 -> d_out

    // --- classifier (2 -> 8) ---
    k_classifier<<<gN, BLK, 0, stream>>>(h3, Wc, bc, out, N);
}